// ATSSAssigner_78537771974789
// MI455X (gfx1250) — compile-verified
//
#include <hip/hip_runtime.h>
#include <hip/hip_bf16.h>
#include <math.h>

#define BS       16
#define NMAX     32
#define A_TOTAL  33600
#define TOPKN    9
#define NLVL     3
#define NCLS     80
#define BGIDX    80

typedef __attribute__((ext_vector_type(2))) float v2f;
typedef __attribute__((ext_vector_type(8))) float v8f;

#if defined(__HIP_DEVICE_COMPILE__) && __has_builtin(__builtin_amdgcn_wmma_f32_16x16x4_f32)
#define USE_WMMA_DIST 1
#else
#define USE_WMMA_DIST 0
#endif

__device__ __forceinline__ float iou4(float4 g, float4 a) {
    float lx = fmaxf(g.x, a.x), ly = fmaxf(g.y, a.y);
    float rx = fminf(g.z, a.z), ry = fminf(g.w, a.w);
    float w = fmaxf(rx - lx, 0.f), h = fmaxf(ry - ly, 0.f);
    float inter = w * h;
    float ag = (g.z - g.x) * (g.w - g.y);
    float aa = (a.z - a.x) * (a.w - a.y);
    return inter / (ag + aa - inter + 1e-9f);
}

// ---------------------------------------------------------------------------
// K1: per (b, level) block: top-9 smallest center-distances per GT.
// Distance^2 via WMMA f32 16x16x4: D = A(16x4) x B(4x16) + C
//   A row m  = [gcx_m, gcy_m, 1, 0]
//   B col n  = [-2*acx_n, -2*acy_n, acx^2+acy^2, 0]
//   C[m][n]  = gcx_m^2 + gcy_m^2        => D = |gc - ac|^2
// ---------------------------------------------------------------------------
__global__ __launch_bounds__(256) void k1_topk(
    const float4* __restrict__ anc4,       // A x 4
    const float4* __restrict__ gtb4,       // bs*nmax x 4
    const float*  __restrict__ mask_gt,    // bs*nmax
    int*          __restrict__ cand_idx)   // bs*nmax*27
{
    constexpr int lvl_start_c[NLVL] = {0, 25600, 32000};
    constexpr int lvl_len_c[NLVL]   = {25600, 6400, 1600};

    const int blk = blockIdx.x;            // bs * 3
    const int b   = blk / NLVL;
    const int lvl = blk % NLVL;
    const int start  = lvl_start_c[lvl];
    const int ntiles = lvl_len_c[lvl] >> 4;   // all levels divisible by 16

    __shared__ float s_gcx[NMAX], s_gcy[NMAX], s_gnorm[NMAX], s_mask[NMAX];
    __shared__ float s_dist[8][NMAX][16];                    // 16 KB
    __shared__ unsigned long long s_top[8][NMAX][TOPKN];     // 18 KB

    const int tid = threadIdx.x;
    if (tid < NMAX) {
        float4 g = gtb4[b * NMAX + tid];
        float gx = (g.x + g.z) * 0.5f, gy = (g.y + g.w) * 0.5f;
        s_gcx[tid] = gx; s_gcy[tid] = gy; s_gnorm[tid] = gx * gx + gy * gy;
        s_mask[tid] = mask_gt[b * NMAX + tid];
    }
    __syncthreads();

    const int wave = tid >> 5;     // wave32: 8 waves
    const int lane = tid & 31;
    const int lh   = lane >> 4;    // lane half
    const int n16  = lane & 15;

    unsigned long long top[TOPKN];
#pragma unroll
    for (int i = 0; i < TOPKN; i++) top[i] = ~0ULL;

#if USE_WMMA_DIST
    v2f amat;
    if (lh == 0) { amat.x = s_gcx[n16]; amat.y = s_gcy[n16]; }
    else         { amat.x = 1.0f;       amat.y = 0.0f; }
    v8f cmat;
#pragma unroll
    for (int j = 0; j < 8; j++) cmat[j] = s_gnorm[j + 8 * lh];
#endif

    for (int t = wave; t < ntiles; t += 8) {
        const int abase = start + t * 16;
        float4 ab = anc4[abase + n16];
        float ax = (ab.x + ab.z) * 0.5f, ay = (ab.y + ab.w) * 0.5f;
        float d[8];
#if USE_WMMA_DIST
        v2f bmat;
        if (lh == 0) { bmat.x = -2.0f * ax;      bmat.y = -2.0f * ay; }
        else         { bmat.x = ax * ax + ay * ay; bmat.y = 0.0f; }
        v8f dm = __builtin_amdgcn_wmma_f32_16x16x4_f32(
            false, amat, false, bmat, (short)0, cmat, false, false);
#pragma unroll
        for (int j = 0; j < 8; j++) d[j] = dm[j];
#else
#pragma unroll
        for (int j = 0; j < 8; j++) {
            int m = j + 8 * lh;
            float dx = s_gcx[m] - ax, dy = s_gcy[m] - ay;
            d[j] = dx * dx + dy * dy;
        }
#endif
        // transpose tile through LDS: row = gt, col = anchor-in-tile
#pragma unroll
        for (int j = 0; j < 8; j++) s_dist[wave][j + 8 * lh][n16] = d[j];
#if __has_builtin(__builtin_amdgcn_wave_barrier)
        __builtin_amdgcn_wave_barrier();   // wave-local: DS ops in-order per wave
#endif
        // each lane owns one gt row; insert 16 candidates
        {
            const int m = lane;
            const float mk = s_mask[m];
#pragma unroll 4
            for (int n = 0; n < 16; n++) {
                float dd = fmaxf(s_dist[wave][m][n], 0.0f) * mk;
                unsigned long long key =
                    ((unsigned long long)__float_as_uint(dd) << 32) |
                    (unsigned)(abase + n);
                if (key < top[TOPKN - 1]) {
                    int p = TOPKN - 1;
                    while (p > 0 && top[p - 1] > key) { top[p] = top[p - 1]; --p; }
                    top[p] = key;
                }
            }
        }
#if __has_builtin(__builtin_amdgcn_wave_barrier)
        __builtin_amdgcn_wave_barrier();
#endif
    }

    for (int i = 0; i < TOPKN; i++) s_top[wave][lane][i] = top[i];
    __syncthreads();

    if (tid < NMAX) {
        const int m = tid;
        unsigned long long fin[TOPKN];
#pragma unroll
        for (int i = 0; i < TOPKN; i++) fin[i] = s_top[0][m][i];
        for (int w = 1; w < 8; w++) {
#pragma unroll
            for (int i = 0; i < TOPKN; i++) {
                unsigned long long key = s_top[w][m][i];
                if (key < fin[TOPKN - 1]) {
                    int p = TOPKN - 1;
                    while (p > 0 && fin[p - 1] > key) { fin[p] = fin[p - 1]; --p; }
                    fin[p] = key;
                } else break;   // lists are sorted ascending
            }
        }
        int* outp = cand_idx + ((b * NMAX + m) * NLVL + lvl) * TOPKN;
        for (int i = 0; i < TOPKN; i++)
            outp[i] = (int)(unsigned)(fin[i] & 0xffffffffu);
    }
}

// ---------------------------------------------------------------------------
// K2: scatter counts (+ owner gt; only consumed when count==1 -> 1 writer)
// ---------------------------------------------------------------------------
__global__ void k2_count(const int* __restrict__ cand_idx,
                         int* __restrict__ cnt, int* __restrict__ owner)
{
    int i = blockIdx.x * blockDim.x + threadIdx.x;   // bs*nmax*27
    if (i >= BS * NMAX * NLVL * TOPKN) return;
    int b  = i / (NMAX * NLVL * TOPKN);
    int gt = (i / (NLVL * TOPKN)) % NMAX;
    int a  = cand_idx[i];
    atomicAdd(&cnt[b * A_TOTAL + a], 1);
    owner[b * A_TOTAL + a] = gt;
}

// ---------------------------------------------------------------------------
// K3: per anchor resolve unique candidate gt + its IoU
// ---------------------------------------------------------------------------
__global__ __launch_bounds__(256) void k3_resolve(
    const int* __restrict__ cnt, const int* __restrict__ owner,
    const float4* __restrict__ anc4, const float4* __restrict__ gtb4,
    int* __restrict__ agt, float* __restrict__ aiou)
{
    int idx = blockIdx.x * blockDim.x + threadIdx.x;   // bs*A
    if (idx >= BS * A_TOTAL) return;
    int b = idx / A_TOTAL, a = idx - b * A_TOTAL;
    int c = cnt[idx];
    int g = -1; float iou = 0.f;
    if (c == 1) {
        g = owner[idx];
        iou = iou4(gtb4[b * NMAX + g], anc4[a]);
    } else if (c > 1) {
        float4 ab = anc4[a];
        float best = -1.f; int bg = 0;
        for (int gi = 0; gi < NMAX; gi++) {
            float v = iou4(gtb4[b * NMAX + gi], ab);
            if (v > best) { best = v; bg = gi; }   // ties -> lowest gt
        }
        g = bg; iou = best;
    }
    agt[idx] = g; aiou[idx] = iou;
}

// ---------------------------------------------------------------------------
// K3b: per (b,gt) threshold = mean + std(ddof=1, over full A axis)
// ---------------------------------------------------------------------------
__global__ __launch_bounds__(256) void k3b_thresh(
    const int* __restrict__ agt, const float* __restrict__ aiou,
    float* __restrict__ thr)
{
    const int b = blockIdx.x / NMAX, g = blockIdx.x % NMAX;
    const int tid = threadIdx.x;
    float s = 0.f, q = 0.f; int c = 0;
    const int base = b * A_TOTAL;
    for (int a = tid; a < A_TOTAL; a += 256) {
        if (agt[base + a] == g) {
            float v = aiou[base + a];
            s += v; q += v * v; c++;
        }
    }
    __shared__ float rs[256], rq[256];
    __shared__ int rc[256];
    rs[tid] = s; rq[tid] = q; rc[tid] = c;
    __syncthreads();
    for (int st = 128; st > 0; st >>= 1) {
        if (tid < st) { rs[tid] += rs[tid + st]; rq[tid] += rq[tid + st]; rc[tid] += rc[tid + st]; }
        __syncthreads();
    }
    if (tid == 0) {
        float S = rs[0], Q = rq[0]; int n = rc[0];
        float t;
        if (n == 0) t = 3.4e38f;                       // NaN-compare-false equivalent
        else {
            float avg = S / (float)n;
            float var = (Q - S * S / (float)A_TOTAL) / (float)(A_TOTAL - 1);
            t = avg + sqrtf(fmaxf(var, 0.f));
        }
        thr[b * NMAX + g] = t;
    }
}

// ---------------------------------------------------------------------------
// K4: final assignment + outputs (scores region pre-zeroed via memsetAsync)
// ---------------------------------------------------------------------------
__global__ __launch_bounds__(256) void k4_assign(
    const int* __restrict__ agt, const float* __restrict__ aiou,
    const float* __restrict__ thr,
    const int* __restrict__ gt_labels, const float4* __restrict__ gtb4,
    const float* __restrict__ mask_gt, const float4* __restrict__ pdb4,
    float* __restrict__ out_labels, float4* __restrict__ out_boxes,
    float* __restrict__ out_scores, float* __restrict__ out_pos)
{
    int idx = blockIdx.x * blockDim.x + threadIdx.x;   // bs*A
    if (idx >= BS * A_TOTAL) return;
    int b = idx / A_TOTAL;

    int g = agt[idx];
    float iou = aiou[idx];
    bool pos = (g >= 0) &&
               (mask_gt[b * NMAX + g] > 0.f) &&
               (iou > thr[b * NMAX + g]);
    int ag = pos ? g : 0;

    out_labels[idx] = pos ? (float)gt_labels[b * NMAX + ag] : (float)BGIDX;
    out_boxes[idx]  = gtb4[b * NMAX + ag];
    out_pos[idx]    = pos ? 1.f : 0.f;

    if (pos) {
        float4 pb = pdb4[idx];
        float best = 0.f;                      // IoUs are >= 0
        for (int gi = 0; gi < NMAX; gi++)
            best = fmaxf(best, iou4(pb, gtb4[b * NMAX + gi]));
        int lbl = gt_labels[b * NMAX + g];     // in [0, 80)
        out_scores[(size_t)idx * NCLS + lbl] = best;
    }
}

// ---------------------------------------------------------------------------
extern "C" void kernel_launch(void* const* d_in, const int* in_sizes, int n_in,
                              void* d_out, int out_size, void* d_ws, size_t ws_size,
                              hipStream_t stream) {
    (void)in_sizes; (void)n_in; (void)out_size; (void)ws_size;

    const float4* anc4   = (const float4*)d_in[0];
    // d_in[1] = n_level_bboxes (int64) -- fixed [25600,6400,1600], hardcoded
    const int*    gt_lab = (const int*)d_in[2];
    const float4* gtb4   = (const float4*)d_in[3];
    const float*  maskg  = (const float*)d_in[4];
    const float4* pdb4   = (const float4*)d_in[5];

    float* out = (float*)d_out;
    const size_t NA = (size_t)BS * A_TOTAL;      // 537600
    float*  out_labels = out;
    float4* out_boxes  = (float4*)(out + NA);
    float*  out_scores = out + NA * 5;
    float*  out_pos    = out + NA * 85;

    auto up = [](size_t x) { return (x + 255) & ~(size_t)255; };
    char* w = (char*)d_ws;
    int*   cand  = (int*)w;                w += up((size_t)BS * NMAX * NLVL * TOPKN * 4);
    int*   cnt   = (int*)w;                w += up(NA * 4);
    int*   owner = (int*)w;                w += up(NA * 4);
    int*   agt   = (int*)w;                w += up(NA * 4);
    float* aiou  = (float*)w;              w += up(NA * 4);
    float* thr   = (float*)w;              w += up((size_t)BS * NMAX * 4);

    hipMemsetAsync(cnt, 0, NA * 4, stream);
    hipMemsetAsync(out_scores, 0, NA * (size_t)NCLS * sizeof(float), stream);

    k1_topk  <<<BS * NLVL, 256, 0, stream>>>(anc4, gtb4, maskg, cand);
    k2_count <<<(BS * NMAX * NLVL * TOPKN + 255) / 256, 256, 0, stream>>>(cand, cnt, owner);
    k3_resolve<<<(int)((NA + 255) / 256), 256, 0, stream>>>(cnt, owner, anc4, gtb4, agt, aiou);
    k3b_thresh<<<BS * NMAX, 256, 0, stream>>>(agt, aiou, thr);
    k4_assign<<<(int)((NA + 255) / 256), 256, 0, stream>>>(
        agt, aiou, thr, gt_lab, gtb4, maskg, pdb4,
        out_labels, out_boxes, out_scores, out_pos);
}